// PairEmbed_23905787969896
// MI455X (gfx1250) — compile-verified
//
#include <hip/hip_runtime.h>
#include <math.h>

// ---------------------------------------------------------------------------
// PairEmbed on MI455X (gfx1250, wave32).
// Memory-bound pipeline; GEMMs done with fp32 WMMA (V_WMMA_F32_16X16X4_F32).
// ---------------------------------------------------------------------------

typedef __attribute__((ext_vector_type(2))) float v2f;
typedef __attribute__((ext_vector_type(8))) float v8f;

#define BB   32
#define SS   256
#define PP   32640            // S*(S-1)/2
#define NT   (PP / 16)        // 2040 N-tiles, exact
#define PIF  3.14159265358979323846f
#define INV_SQRT2 0.70710678118654752f

__device__ __forceinline__ float gelu_exact(float v) {
    return 0.5f * v * (1.0f + erff(v * INV_SQRT2));
}

// decode pair index p -> (i, j) with i > j, tril_indices(-1) row-major order
__device__ __forceinline__ void decode_pair(int p, int& i, int& j) {
    i = (int)((1.0f + sqrtf(8.0f * (float)p + 1.0f)) * 0.5f);
    while (i * (i - 1) / 2 > p) --i;
    while ((i + 1) * i / 2 <= p) ++i;
    j = p - i * (i - 1) / 2;
}

// ---------------------------------------------------------------------------
// K0: zero small scratch (BN stat accumulators)
// ---------------------------------------------------------------------------
__global__ void zero_kernel(float* __restrict__ p, int n) {
    int t = blockIdx.x * blockDim.x + threadIdx.x;
    if (t < n) p[t] = 0.0f;
}

// ---------------------------------------------------------------------------
// K1: pairwise lorentz-vector features -> h0 (B,4,P), accumulate BN0 stats
// ---------------------------------------------------------------------------
__global__ void feat_kernel(const float* __restrict__ x,
                            float* __restrict__ h0,
                            float* __restrict__ stats0) {  // [4 sums | 4 sumsq]
    __shared__ float ls[8];
    if (threadIdx.x < 8) ls[threadIdx.x] = 0.0f;
    __syncthreads();

    int gid = blockIdx.x * blockDim.x + threadIdx.x;
    float vals[4] = {0.f, 0.f, 0.f, 0.f};
    bool active = gid < BB * PP;
    if (active) {
        int b = gid / PP;
        int p = gid - b * PP;
        int i, j; decode_pair(p, i, j);
        const float* xb = x + (size_t)b * 4 * SS;
        float pxi = xb[0*SS+i], pyi = xb[1*SS+i], pzi = xb[2*SS+i], ei = xb[3*SS+i];
        float pxj = xb[0*SS+j], pyj = xb[1*SS+j], pzj = xb[2*SS+j], ej = xb[3*SS+j];

        float pti  = sqrtf(pxi*pxi + pyi*pyi);
        float ptj  = sqrtf(pxj*pxj + pyj*pyj);
        float rapi = 0.5f * log1pf(2.0f*pzi / fmaxf(ei - pzi, 1e-20f));
        float rapj = 0.5f * log1pf(2.0f*pzj / fmaxf(ej - pzj, 1e-20f));
        float phii = atan2f(pyi, pxi);
        float phij = atan2f(pyj, pxj);

        float dphi = fmodf(phii - phij + PIF, 2.0f*PIF);
        if (dphi < 0.0f) dphi += 2.0f*PIF;   // python-style mod
        dphi -= PIF;

        float dr    = rapi - rapj;
        float delta = sqrtf(dr*dr + dphi*dphi);
        float lndelta = log1pf(fmaxf(delta, 1e-8f));
        float ptmin   = fminf(pti, ptj);
        float lnkt    = log1pf(fmaxf(ptmin * delta, 1e-8f));
        float lnz     = log1pf(fmaxf(ptmin / fmaxf(pti + ptj, 1e-8f), 1e-8f));
        float sx = pxi+pxj, sy = pyi+pyj, sz = pzi+pzj, se = ei+ej;
        float m2 = fmaxf(se*se - (sx*sx + sy*sy + sz*sz), 1e-8f);
        float lnm2 = log1pf(m2);
        float mask = (pti != 0.0f && ptj != 0.0f) ? 1.0f : 0.0f;

        vals[0] = lnkt * mask;
        vals[1] = lnz * mask;
        vals[2] = lndelta * mask;
        vals[3] = lnm2 * mask;

        float* hb = h0 + (size_t)b * 4 * PP + p;
        hb[0]        = vals[0];
        hb[PP]       = vals[1];
        hb[2*(size_t)PP] = vals[2];
        hb[3*(size_t)PP] = vals[3];
    }
    if (active) {
        #pragma unroll
        for (int c = 0; c < 4; ++c) {
            atomicAdd(&ls[c],     vals[c]);
            atomicAdd(&ls[4 + c], vals[c] * vals[c]);
        }
    }
    __syncthreads();
    if (threadIdx.x < 8) atomicAdd(&stats0[threadIdx.x], ls[threadIdx.x]);
}

// ---------------------------------------------------------------------------
// K2: finalize BN stats: raw [sum|sumsq] -> murs [mu|rsqrt(var+eps)]
// ---------------------------------------------------------------------------
__global__ void finalize_stats(const float* __restrict__ raw,
                               float* __restrict__ murs, int C, float inv_n) {
    int c = blockIdx.x * blockDim.x + threadIdx.x;
    if (c < C) {
        float mu  = raw[c] * inv_n;
        float var = raw[C + c] * inv_n - mu * mu;
        murs[c]     = mu;
        murs[C + c] = rsqrtf(var + 1e-5f);
    }
}

// ---------------------------------------------------------------------------
// Layer kernel: BN + GELU + GEMM (W: COUT x CIN applied to h: CIN x P per batch)
// One wave32 per (batch, 16-pair tile); fp32 WMMA 16x16x4, K-loop over CIN.
// FINAL=true scatters symmetric into (B,COUT,S,S) output instead of (B,COUT,P).
// ---------------------------------------------------------------------------
template <int CIN, int COUT, bool ACC_STATS, bool FINAL>
__global__ void layer_kernel(const float* __restrict__ hin,
                             const float* __restrict__ murs,   // [CIN mu | CIN rs]
                             const float* __restrict__ gamma,
                             const float* __restrict__ beta,
                             const float* __restrict__ w,      // COUT x CIN row-major
                             const float* __restrict__ cbias,  // COUT
                             float* __restrict__ hout,
                             float* __restrict__ stats_next) { // [COUT sums | COUT sumsq]
    constexpr int MT = COUT / 16;
    __shared__ float lsum[COUT];
    __shared__ float lsq[COUT];
    if (ACC_STATS) {
        for (int t = threadIdx.x; t < COUT; t += blockDim.x) { lsum[t] = 0.f; lsq[t] = 0.f; }
        __syncthreads();
    }

    const int wave = blockIdx.x * (blockDim.x >> 5) + (threadIdx.x >> 5);
    const int lane = threadIdx.x & 31;
    const int b  = wave / NT;          // grid sized so every wave is active (EXEC all-1s)
    const int nt = wave - b * NT;
    const int p0 = nt * 16;
    const int n  = lane & 15;          // N index within tile (B/C/D frags)
    const int kb = (lane >> 4) * 2;    // K sub-offset for A/B frags (ISA 16x4 layout)

    v8f acc[MT];
    const v8f vzero = {0.f, 0.f, 0.f, 0.f, 0.f, 0.f, 0.f, 0.f};
    #pragma unroll
    for (int mt = 0; mt < MT; ++mt) acc[mt] = vzero;

    const float* hb = hin + (size_t)b * CIN * PP;
    #pragma unroll
    for (int k0 = 0; k0 < CIN; k0 += 4) {
        const int c0 = k0 + kb, c1 = c0 + 1;
        float v0 = hb[(size_t)c0 * PP + p0 + n];
        float v1 = hb[(size_t)c1 * PP + p0 + n];
        // batchnorm affine + exact GELU fused into the B-fragment load
        v0 = (v0 - murs[c0]) * murs[CIN + c0] * gamma[c0] + beta[c0];
        v1 = (v1 - murs[c1]) * murs[CIN + c1] * gamma[c1] + beta[c1];
        v0 = gelu_exact(v0);
        v1 = gelu_exact(v1);
        v2f bf = {v0, v1};
        #pragma unroll
        for (int mt = 0; mt < MT; ++mt) {
            const float* wr = w + (size_t)(mt * 16 + (lane & 15)) * CIN + c0;
            v2f af = {wr[0], wr[1]};
            acc[mt] = __builtin_amdgcn_wmma_f32_16x16x4_f32(
                /*neg_a=*/false, af, /*neg_b=*/false, bf,
                /*c_mod=*/(short)0, acc[mt], /*reuse_a=*/false, /*reuse_b=*/false);
        }
    }

    // writeback: C/D layout -> VGPR r holds M = r + 8*(lane>=16), N = lane&15
    const int mhi = (lane >> 4) << 3;
    #pragma unroll
    for (int mt = 0; mt < MT; ++mt) {
        #pragma unroll
        for (int r = 0; r < 8; ++r) {
            const int ch = mt * 16 + r + mhi;
            float val = acc[mt][r] + cbias[ch];
            if (!FINAL) {
                hout[(size_t)b * COUT * PP + (size_t)ch * PP + p0 + n] = val;
            } else {
                int p = p0 + n, i, j;
                decode_pair(p, i, j);
                float* yb = hout + ((size_t)b * COUT + ch) * SS * SS;
                yb[i * SS + j] = val;
                yb[j * SS + i] = val;
            }
            if (ACC_STATS) {
                atomicAdd(&lsum[ch], val);
                atomicAdd(&lsq[ch], val * val);
            }
        }
    }
    if (ACC_STATS) {
        __syncthreads();
        for (int t = threadIdx.x; t < COUT; t += blockDim.x) {
            atomicAdd(&stats_next[t],        lsum[t]);
            atomicAdd(&stats_next[COUT + t], lsq[t]);
        }
    }
}

// ---------------------------------------------------------------------------
// zero the output diagonal (only positions never written by the scatter)
// ---------------------------------------------------------------------------
__global__ void diag_kernel(float* __restrict__ y) {
    int t = blockIdx.x * blockDim.x + threadIdx.x;
    if (t < BB * 16 * SS) {
        int s  = t % SS;
        int bc = t / SS;
        y[(size_t)bc * SS * SS + (size_t)s * (SS + 1)] = 0.0f;
    }
}

// ---------------------------------------------------------------------------
extern "C" void kernel_launch(void* const* d_in, const int* in_sizes, int n_in,
                              void* d_out, int out_size, void* d_ws, size_t ws_size,
                              hipStream_t stream) {
    const float* x  = (const float*)d_in[0];
    const float* g0 = (const float*)d_in[1];
    const float* be0= (const float*)d_in[2];
    const float* w0 = (const float*)d_in[3];
    const float* c0 = (const float*)d_in[4];
    const float* g1 = (const float*)d_in[5];
    const float* be1= (const float*)d_in[6];
    const float* w1 = (const float*)d_in[7];
    const float* c1 = (const float*)d_in[8];
    const float* g2 = (const float*)d_in[9];
    const float* be2= (const float*)d_in[10];
    const float* w2 = (const float*)d_in[11];
    const float* c2 = (const float*)d_in[12];
    float* out = (float*)d_out;

    // workspace layout (floats): h0 | h1 | h2 | raw0(8) raw1(128) raw2(128) | murs0(8) murs1(128) murs2(128)
    const size_t H0 = (size_t)BB * 4 * PP;
    const size_t H1 = (size_t)BB * 64 * PP;
    float* ws    = (float*)d_ws;
    float* h0    = ws;
    float* h1    = ws + H0;
    float* h2    = ws + H0 + H1;
    float* raw0  = ws + H0 + 2 * H1;
    float* raw1  = raw0 + 8;
    float* raw2  = raw1 + 128;
    float* murs0 = raw2 + 128;
    float* murs1 = murs0 + 8;
    float* murs2 = murs1 + 128;

    const float inv_n = 1.0f / (float)((size_t)BB * PP);

    // K0: zero stat scratch (528 floats)
    zero_kernel<<<3, 256, 0, stream>>>(raw0, 528);

    // K1: features + BN0 stats
    feat_kernel<<<(BB * PP + 255) / 256, 256, 0, stream>>>(x, h0, raw0);

    // K2
    finalize_stats<<<1, 32, 0, stream>>>(raw0, murs0, 4, inv_n);

    // K3: layer0 (4 -> 64), accumulate BN1 stats.  waves = B*NT = 65280 -> 8160 blocks of 8 waves
    layer_kernel<4, 64, true, false><<<(BB * NT) / 8, 256, 0, stream>>>(
        h0, murs0, g0, be0, w0, c0, h1, raw1);

    // K4
    finalize_stats<<<2, 64, 0, stream>>>(raw1, murs1, 64, inv_n);

    // K5: layer1 (64 -> 64), accumulate BN2 stats
    layer_kernel<64, 64, true, false><<<(BB * NT) / 8, 256, 0, stream>>>(
        h1, murs1, g1, be1, w1, c1, h2, raw2);

    // K6
    finalize_stats<<<2, 64, 0, stream>>>(raw2, murs2, 64, inv_n);

    // K7: zero diagonal, then layer2 (64 -> 16) fused with symmetric scatter
    diag_kernel<<<(BB * 16 * SS + 255) / 256, 256, 0, stream>>>(out);
    layer_kernel<64, 16, false, true><<<(BB * NT) / 8, 256, 0, stream>>>(
        h2, murs2, g2, be2, w2, c2, out, nullptr);
}